// TemporalGNN_11665131176208
// MI455X (gfx1250) — compile-verified
//
#include <hip/hip_runtime.h>

#define N_NODES 50000
#define N_EDGES 1600000
#define T 12
// F_IN=4, F_OUT=32; Y layout: [node][f(4)][t(12)] -> 48 floats/node

typedef __attribute__((ext_vector_type(2))) float v2f;
typedef __attribute__((ext_vector_type(8))) float v8f;

// ---------------- degree / norm ----------------
__global__ void deg_init(float* __restrict__ deg) {
  int n = blockIdx.x * blockDim.x + threadIdx.x;
  if (n < N_NODES) deg[n] = 1.0f;  // self loop
}

__global__ void deg_count(const int* __restrict__ ei, float* __restrict__ deg) {
  int e = blockIdx.x * blockDim.x + threadIdx.x;
  if (e < N_EDGES) atomicAdd(&deg[ei[N_EDGES + e]], 1.0f);  // count by target (col)
}

// dinv = rsqrt(deg); Y = dinv^2 * x  (self-loop contribution)
__global__ void selfloop_init(const float* __restrict__ x, const float* __restrict__ deg,
                              float* __restrict__ dinv, float* __restrict__ Y) {
  int n = blockIdx.x * blockDim.x + threadIdx.x;
  if (n >= N_NODES) return;
  float dv = rsqrtf(deg[n]);
  dinv[n] = dv;
  float w = dv * dv;
  const float4* xs = (const float4*)(x + (size_t)n * 48);
  float4* yd = (float4*)(Y + (size_t)n * 48);
#pragma unroll
  for (int q = 0; q < 12; ++q) {
    float4 v = xs[q];
    yd[q] = make_float4(w * v.x, w * v.y, w * v.z, w * v.w);
  }
}

// ---------------- SpMM scatter: Y[col] += dinv[row]*dinv[col] * x[row] ----------------
__global__ void edge_scatter(const float* __restrict__ x, const int* __restrict__ ei,
                             const float* __restrict__ dinv, float* __restrict__ Y) {
  int e = blockIdx.x * blockDim.x + threadIdx.x;
  if (e >= N_EDGES) return;
  int s = ei[e];             // row (source)
  int d = ei[N_EDGES + e];   // col (target)
  float w = dinv[s] * dinv[d];
  const float4* xs = (const float4*)(x + (size_t)s * 48);
  float* yd = Y + (size_t)d * 48;
#pragma unroll
  for (int q = 0; q < 12; ++q) {
    float4 v = xs[q];
    atomicAdd(yd + 4 * q + 0, w * v.x);
    atomicAdd(yd + 4 * q + 1, w * v.y);
    atomicAdd(yd + 4 * q + 2, w * v.z);
    atomicAdd(yd + 4 * q + 3, w * v.w);
  }
}

// ---------------- fold params: Mz = Wz @ Lz[:32], cz = bz@Lz[:32]+lz; softmax(att) ----------------
__global__ void precompute(const float* __restrict__ Wz, const float* __restrict__ bz,
                           const float* __restrict__ Lz, const float* __restrict__ lz,
                           const float* __restrict__ Wh, const float* __restrict__ bh,
                           const float* __restrict__ Lh, const float* __restrict__ lh,
                           const float* __restrict__ att,
                           float* __restrict__ Mz, float* __restrict__ cz,
                           float* __restrict__ Mh, float* __restrict__ ch,
                           float* __restrict__ probs) {
  int tid = threadIdx.x;  // 128 threads
  if (tid < 128) {
    int f = tid >> 5, c = tid & 31;
    float sz = 0.f, sh = 0.f;
    for (int j = 0; j < 32; ++j) {
      sz += Wz[f * 32 + j] * Lz[j * 32 + c];   // Lz is (64,32); top half rows 0..31
      sh += Wh[f * 32 + j] * Lh[j * 32 + c];
    }
    Mz[tid] = sz;
    Mh[tid] = sh;
  }
  if (tid < 32) {
    float sz = lz[tid], sh = lh[tid];
    for (int j = 0; j < 32; ++j) {
      sz += bz[j] * Lz[j * 32 + tid];
      sh += bh[j] * Lh[j * 32 + tid];
    }
    cz[tid] = sz;
    ch[tid] = sh;
  }
  if (tid == 0) {
    float a[T], mx = -3.0e38f;
    for (int t = 0; t < T; ++t) { a[t] = att[t]; mx = fmaxf(mx, a[t]); }
    float s = 0.f;
    for (int t = 0; t < T; ++t) { a[t] = __expf(a[t] - mx); s += a[t]; }
    for (int t = 0; t < T; ++t) probs[t] = a[t] / s;
  }
}

__device__ inline float sigmoid_f(float x) { return 1.0f / (1.0f + __expf(-x)); }
__device__ inline float tanh_f(float x) {
  float e = __expf(2.0f * x);            // inf-safe: 1 - 2/(e+1) -> +1 / -1 at extremes
  return 1.0f - 2.0f / (e + 1.0f);
}

// ---------------- dense stage with V_WMMA_F32_16X16X4_F32 ----------------
// One wave32 per 16-node tile. A = Y-tile (16x4, K=f), B = Mz/Mh column chunks (4x16).
__global__ __launch_bounds__(128) void dense_wmma(
    const float* __restrict__ Y, const float* __restrict__ Mz, const float* __restrict__ cz,
    const float* __restrict__ Mh, const float* __restrict__ ch, const float* __restrict__ probs,
    const float* __restrict__ Wo, const float* __restrict__ bo, float* __restrict__ out) {
  __shared__ float lds[4 * 16 * 32];  // per-wave 16 nodes x 32 channels
  const int NT = N_NODES / 16;        // 3125 tiles
  int wave = threadIdx.x >> 5;
  int lane = threadIdx.x & 31;
  int tile = blockIdx.x * 4 + wave;
  if (tile >= NT) return;             // wave-uniform exit; active waves keep EXEC all-1

  int m = lane & 15;                  // M row (A) / N col (B,D)
  int kh = lane >> 4;                 // lane half selects K pair {0,1} vs {2,3}
  int nodeBase = tile * 16;

  // A fragments: lane m holds A[m, 2kh] (v0) and A[m, 2kh+1] (v1); per period t.
  // Y offset (node,f,t) = node*48 + f*12 + t ; f = 2*kh -> base kh*24.
  const float4* y4 = (const float4*)(Y + (size_t)(nodeBase + m) * 48 + kh * 24);
  float ya[T], yb[T];
#pragma unroll
  for (int q = 0; q < 3; ++q) {
    float4 v = y4[q];
    ya[4 * q] = v.x; ya[4 * q + 1] = v.y; ya[4 * q + 2] = v.z; ya[4 * q + 3] = v.w;
  }
#pragma unroll
  for (int q = 0; q < 3; ++q) {
    float4 v = y4[3 + q];
    yb[4 * q] = v.x; yb[4 * q + 1] = v.y; yb[4 * q + 2] = v.z; yb[4 * q + 3] = v.w;
  }

  // B fragments (4x16): lane n holds B[koff+0,n] (v0), B[koff+1,n] (v1), koff = 2*kh.
  int koff = kh * 2;
  v2f Bzlo, Bzhi, Bhlo, Bhhi;
  Bzlo[0] = Mz[koff * 32 + m];          Bzlo[1] = Mz[(koff + 1) * 32 + m];
  Bzhi[0] = Mz[koff * 32 + 16 + m];     Bzhi[1] = Mz[(koff + 1) * 32 + 16 + m];
  Bhlo[0] = Mh[koff * 32 + m];          Bhlo[1] = Mh[(koff + 1) * 32 + m];
  Bhhi[0] = Mh[koff * 32 + 16 + m];     Bhhi[1] = Mh[(koff + 1) * 32 + 16 + m];

  float czlo = cz[m], czhi = cz[16 + m];
  float chlo = ch[m], chhi = ch[16 + m];
  float pr[T];
#pragma unroll
  for (int t = 0; t < T; ++t) pr[t] = probs[t];

  v8f accLo = {};  // channels 0..15  (D: vgpr v -> node kh*8+v, lane n -> chan n)
  v8f accHi = {};  // channels 16..31

#pragma unroll
  for (int t = 0; t < T; ++t) {
    v2f a; a[0] = ya[t]; a[1] = yb[t];
    v8f zero = {};
    v8f dzl = __builtin_amdgcn_wmma_f32_16x16x4_f32(false, a, false, Bzlo, (short)0, zero, false, false);
    v8f dzh = __builtin_amdgcn_wmma_f32_16x16x4_f32(false, a, false, Bzhi, (short)0, zero, false, false);
    v8f dhl = __builtin_amdgcn_wmma_f32_16x16x4_f32(false, a, false, Bhlo, (short)0, zero, false, false);
    v8f dhh = __builtin_amdgcn_wmma_f32_16x16x4_f32(false, a, false, Bhhi, (short)0, zero, false, false);
    float p = pr[t];
#pragma unroll
    for (int i = 0; i < 8; ++i) {
      float z = sigmoid_f(dzl[i] + czlo);
      float h = tanh_f(dhl[i] + chlo);
      accLo[i] += p * (1.0f - z) * h;
      z = sigmoid_f(dzh[i] + czhi);
      h = tanh_f(dhh[i] + chhi);
      accHi[i] += p * (1.0f - z) * h;
    }
  }

  // relu + transpose via per-wave LDS slice: lds[node][chan]
  float* my = lds + wave * 512;
#pragma unroll
  for (int v = 0; v < 8; ++v) {
    int r = kh * 8 + v;  // node row within tile
    my[r * 32 + m] = fmaxf(accLo[v], 0.0f);
    my[r * 32 + 16 + m] = fmaxf(accHi[v], 0.0f);
  }
  asm volatile("s_wait_dscnt 0" ::: "memory");  // wave-local DS ordering (CDNA5 split counter)

  // final projection relu(H) @ Wo + bo : 192 outputs -> 6 per lane
#pragma unroll
  for (int j = 0; j < 6; ++j) {
    int idx = lane * 6 + j;
    int nl = idx / 12, p = idx % 12;
    float s = bo[p];
#pragma unroll
    for (int c = 0; c < 32; ++c) s += my[nl * 32 + c] * Wo[c * 12 + p];
    out[(size_t)(nodeBase + nl) * 12 + p] = s;
  }
}

extern "C" void kernel_launch(void* const* d_in, const int* in_sizes, int n_in,
                              void* d_out, int out_size, void* d_ws, size_t ws_size,
                              hipStream_t stream) {
  (void)in_sizes; (void)n_in; (void)out_size; (void)ws_size;
  const float* x   = (const float*)d_in[0];
  const int*   ei  = (const int*)d_in[1];
  const float* att = (const float*)d_in[2];
  const float* Wz  = (const float*)d_in[3];
  const float* bz  = (const float*)d_in[4];
  const float* Lz  = (const float*)d_in[5];
  const float* lz  = (const float*)d_in[6];
  // d_in[7..10] = Wr, br, Lr, lr : mathematically dead (H0 == 0)
  const float* Wh  = (const float*)d_in[11];
  const float* bh  = (const float*)d_in[12];
  const float* Lh  = (const float*)d_in[13];
  const float* lh  = (const float*)d_in[14];
  const float* Wo  = (const float*)d_in[15];
  const float* bo  = (const float*)d_in[16];
  float* out = (float*)d_out;

  float* ws    = (float*)d_ws;
  float* deg   = ws;
  float* dinv  = ws + N_NODES;
  float* Y     = ws + 2 * N_NODES;                // 48*N floats
  float* Mz    = ws + 2 * N_NODES + 48 * N_NODES; // 50*N
  float* cz    = Mz + 128;
  float* Mh    = cz + 32;
  float* ch    = Mh + 128;
  float* probs = ch + 32;

  deg_init<<<(N_NODES + 255) / 256, 256, 0, stream>>>(deg);
  deg_count<<<(N_EDGES + 255) / 256, 256, 0, stream>>>(ei, deg);
  precompute<<<1, 128, 0, stream>>>(Wz, bz, Lz, lz, Wh, bh, Lh, lh, att, Mz, cz, Mh, ch, probs);
  selfloop_init<<<(N_NODES + 255) / 256, 256, 0, stream>>>(x, deg, dinv, Y);
  edge_scatter<<<(N_EDGES + 255) / 256, 256, 0, stream>>>(x, ei, dinv, Y);

  const int NT = N_NODES / 16;  // 3125
  dense_wmma<<<(NT + 3) / 4, 128, 0, stream>>>(Y, Mz, cz, Mh, ch, probs, Wo, bo, out);
}